// NeighborlistVerletNsq_66486093742353
// MI455X (gfx1250) — compile-verified
//
#include <hip/hip_runtime.h>
#include <stdint.h>

// Problem constants (from the reference)
#define NPART  6144
#define NPAIR  18871296            // NPART*(NPART-1)/2
#define TPB    256
#define CUT    0.5f
#define CUTSK  0.6f                // cutoff + skin

typedef __attribute__((ext_vector_type(4))) unsigned int v4u;
typedef __attribute__((ext_vector_type(8))) int          v8i;
typedef __attribute__((ext_vector_type(4))) int          v4i;

struct __attribute__((aligned(4))) F3 { float x, y, z; };

__global__ __launch_bounds__(TPB)
void nlist_verlet_nsq_kernel(const float* __restrict__ pos,
                             const float* __restrict__ box,
                             float* __restrict__ out)
{
    const int i  = (int)blockIdx.y;            // row index 0..N-2
    const int j0 = (int)blockIdx.x * TPB;      // column chunk base

    // Whole block strictly below/on the diagonal -> nothing to do (uniform exit).
    if (j0 + (TPB - 1) <= i) return;

    __shared__ float shp[TPB * 3];             // 3 KB tile of positions[j0 .. j0+255]

    if (threadIdx.x == 0) {
        // ---- Tensor Data Mover: DMA the 3 KB j-chunk Global -> LDS ----
        // D# group 0 (128b): count=1 | lds_addr | global_addr[56:0] | type=2
        unsigned lds_byte = (unsigned)(uintptr_t)(void*)shp;   // low 32b = LDS offset
        unsigned long long ga =
            (unsigned long long)(uintptr_t)(pos + (size_t)j0 * 3);
        v4u g0;
        g0.x = 1u;                                             // count=1, user mode
        g0.y = lds_byte;                                       // lds_addr (bytes)
        g0.z = (unsigned)(ga & 0xFFFFFFFFu);                   // global_addr[31:0]
        g0.w = (unsigned)((ga >> 32) & 0x1FFFFFFu)             // global_addr[56:32]
             | (2u << 30);                                     // type = 2 ("image")

        // D# group 1 (256b): data_size=4B, 1-D tensor, tile_dim0 = 768 elements
        unsigned remaining = (unsigned)(NPART * 3 - j0 * 3);   // tensor_dim0 (OOB->0)
        v8i g1;
        g1[0] = (int)(2u << 16);                               // data_size=2 (4 bytes)
        g1[1] = (int)((remaining & 0xFFFFu) << 16);            // tensor_dim0[15:0]
        g1[2] = (int)((remaining >> 16) & 0xFFFFu);            // tensor_dim0[31:16]; dim1=0
        g1[3] = (int)((unsigned)(TPB * 3) << 16);              // tile_dim0 = 768
        g1[4] = 0;                                             // tile_dim1=0 (1-D), tile_dim2=0
        g1[5] = 0; g1[6] = 0; g1[7] = 0;                       // strides unused for 1-D tile

        v4i gz4 = {0, 0, 0, 0};                                // groups 2/3: <=2D tensor
        v8i gz8 = {0, 0, 0, 0, 0, 0, 0, 0};
        __builtin_amdgcn_tensor_load_to_lds(g0, g1, gz4, gz4, gz8, 0);
        __builtin_amdgcn_s_wait_tensorcnt(0);                  // DMA complete
    }
    __syncthreads();

    const int j = j0 + (int)threadIdx.x;
    if (j <= i || j >= NPART) return;

    // Row-i position: uniform across the block -> scalar loads, L2-resident.
    const float xi = pos[3 * i + 0];
    const float yi = pos[3 * i + 1];
    const float zi = pos[3 * i + 2];

    // Box lengths (diagonal of box_vectors): uniform scalar loads.
    const float Lx = box[0], Ly = box[4], Lz = box[8];
    const float hx = 0.5f * Lx, hy = 0.5f * Ly, hz = 0.5f * Lz;
    const float ix_ = 1.0f / Lx, iy_ = 1.0f / Ly, iz_ = 1.0f / Lz;

    // Column-j position from LDS (stride-3 across lanes -> bank-conflict free).
    const float xj = shp[3 * threadIdx.x + 0];
    const float yj = shp[3 * threadIdx.x + 1];
    const float zj = shp[3 * threadIdx.x + 2];

    // Minimum image: remainder(d + L/2, L) - L/2  (Python-mod semantics).
    float tx = (xi - xj) + hx; tx -= floorf(tx * ix_) * Lx; const float rx = tx - hx;
    float ty = (yi - yj) + hy; ty -= floorf(ty * iy_) * Ly; const float ry = ty - hy;
    float tz = (zi - zj) + hz; tz -= floorf(tz * iz_) * Lz; const float rz = tz - hz;

    const float d = sqrtf(rx * rx + ry * ry + rz * rz);

    // Triangular pair index: p = i*(2N-i-1)/2 + (j-i-1)   (fits in int32)
    const int p = (i * (2 * NPART - i - 1)) / 2 + (j - i - 1);

    // Outputs, concatenated flat: r_ij[P,3] | d_ij[P] | in_build[P] | in_cutoff[P]
    F3 r3; r3.x = rx; r3.y = ry; r3.z = rz;
    *reinterpret_cast<F3*>(out + (size_t)3 * p) = r3;          // 12B/lane b96 store

    // Dense unit-stride streams: never re-read -> non-temporal (stream past L2).
    __builtin_nontemporal_store(d, out + (size_t)3 * NPAIR + p);
    __builtin_nontemporal_store((d <  CUTSK) ? 1.0f : 0.0f, out + (size_t)4 * NPAIR + p);
    __builtin_nontemporal_store((d <= CUT)   ? 1.0f : 0.0f, out + (size_t)5 * NPAIR + p);
}

extern "C" void kernel_launch(void* const* d_in, const int* in_sizes, int n_in,
                              void* d_out, int out_size, void* d_ws, size_t ws_size,
                              hipStream_t stream) {
    (void)in_sizes; (void)n_in; (void)out_size; (void)d_ws; (void)ws_size;
    const float* pos = (const float*)d_in[0];   // positions [N,3] float32
    const float* box = (const float*)d_in[1];   // box_vectors [3,3] float32
    float* out = (float*)d_out;                 // 6*NPAIR float32

    dim3 grid((NPART + TPB - 1) / TPB, NPART - 1, 1);  // 24 j-chunks x 6143 rows
    dim3 block(TPB, 1, 1);
    nlist_verlet_nsq_kernel<<<grid, block, 0, stream>>>(pos, box, out);
}